// LocalVarianceNet_12481174962725
// MI455X (gfx1250) — compile-verified
//
#include <hip/hip_runtime.h>

typedef __attribute__((ext_vector_type(2))) float v2f;
typedef __attribute__((ext_vector_type(8))) float v8f;

#define WSZ     7
#define PAD     3
#define TILE_H  16
#define TILE_W  128
#define IN_ROWS (TILE_H + WSZ - 1)   // 22
#define IN_COLS (TILE_W + WSZ - 1)   // 134
#define XSTRIDE 136                  // padded LDS row stride for raw tile
#define HROWS   24                   // 22 rounded up to K-chunk multiple of 4
#define IMG_H   512
#define IMG_W   512

// var = Q/49 - (S/49)^2 over a 7x7 circular window.
// Vertical 7-tap sum done as banded-matrix x data-tile via V_WMMA_F32_16X16X4_F32.
__global__ __launch_bounds__(256) void local_variance_kernel(
    const float* __restrict__ x, float* __restrict__ out)
{
    __shared__ float xbuf[IN_ROWS][XSTRIDE];   // raw input tile (with halo)
    __shared__ float hs[HROWS][TILE_W];        // horizontal 7-tap sum of x
    __shared__ float hq[HROWS][TILE_W];        // horizontal 7-tap sum of x^2

    const int tid = threadIdx.x;
    const int img = blockIdx.z;                       // 0..47  (batch*channel)
    const int y0  = blockIdx.y * TILE_H;              // output row base
    const int c0  = blockIdx.x * TILE_W;              // output col base
    const float* __restrict__ src = x + (size_t)img * IMG_H * IMG_W;
    float* __restrict__ dst       = out + (size_t)img * IMG_H * IMG_W;

    // ---- Phase 1: global -> LDS, circular wrap on both axes (coalesced rows)
    for (int idx = tid; idx < IN_ROWS * IN_COLS; idx += 256) {
        int r  = idx / IN_COLS;
        int cc = idx - r * IN_COLS;
        int gy = (y0 + r  - PAD + IMG_H) & (IMG_H - 1);
        int gx = (c0 + cc - PAD + IMG_W) & (IMG_W - 1);
        xbuf[r][cc] = src[gy * IMG_W + gx];
    }
    __syncthreads();

    // ---- Phase 2: horizontal 7-tap sums of x and x^2 into LDS
    for (int idx = tid; idx < IN_ROWS * TILE_W; idx += 256) {
        int r = idx / TILE_W;
        int c = idx - r * TILE_W;
        float s = 0.0f, q = 0.0f;
#pragma unroll
        for (int j = 0; j < WSZ; ++j) {
            float v = xbuf[r][c + j];
            s += v;
            q += v * v;
        }
        hs[r][c] = s;
        hq[r][c] = q;
    }
    // Zero the K-padding rows (22,23): WMMA propagates NaN, pad must be 0.
    for (int idx = tid; idx < (HROWS - IN_ROWS) * TILE_W; idx += 256) {
        int r = IN_ROWS + idx / TILE_W;
        int c = idx % TILE_W;
        hs[r][c] = 0.0f;
        hq[r][c] = 0.0f;
    }
    __syncthreads();

    // ---- Phase 3: vertical 7-tap sum as WMMA (16x22 band matrix x 22x16 tile)
    // Wave w owns the 16-wide column tile starting at w*16 (8 waves, 128 cols).
    const int lane = tid & 31;
    const int wv   = tid >> 5;
    const int n    = lane & 15;       // N (and A-matrix M) for this lane
    const int hf   = lane >> 4;       // 0: lanes 0-15, 1: lanes 16-31
    const int cb   = wv * 16;         // column-tile base within the LDS tile

    v8f accS = {};
    v8f accQ = {};
#pragma unroll
    for (int kc = 0; kc < 6; ++kc) {
        const int k0 = kc * 4 + hf * 2;   // K index held in .x; .y holds k0+1
        // A fragment: band matrix, A[m][k] = 1 iff m <= k <= m+6
        // (output row m consumes input rows m..m+6 of the 22-row strip)
        v2f a;
        a.x = (k0     >= n && k0     <= n + 6) ? 1.0f : 0.0f;
        a.y = (k0 + 1 >= n && k0 + 1 <= n + 6) ? 1.0f : 0.0f;
        // B fragments from LDS: row K, column cb+n (conflict-free: lanes
        // 0-15 read consecutive floats of one row)
        v2f bs, bq;
        bs.x = hs[k0][cb + n];
        bs.y = hs[k0 + 1][cb + n];
        bq.x = hq[k0][cb + n];
        bq.y = hq[k0 + 1][cb + n];
        accS = __builtin_amdgcn_wmma_f32_16x16x4_f32(
            false, a, false, bs, (short)0, accS, false, false);
        accQ = __builtin_amdgcn_wmma_f32_16x16x4_f32(
            false, a, false, bq, (short)0, accQ, false, false);
    }

    // ---- Epilogue: variance from accumulators, coalesced stores.
    // D layout: VGPR v holds rows v (lanes 0-15) and v+8 (lanes 16-31), N=n.
    const float inv_area = 1.0f / (float)(WSZ * WSZ);
    const int gx = c0 + cb + n;
#pragma unroll
    for (int v = 0; v < 8; ++v) {
        int gy = y0 + v + hf * 8;
        float s = accS[v] * inv_area;
        float q = accQ[v] * inv_area;
        dst[gy * IMG_W + gx] = q - s * s;
    }
}

extern "C" void kernel_launch(void* const* d_in, const int* in_sizes, int n_in,
                              void* d_out, int out_size, void* d_ws, size_t ws_size,
                              hipStream_t stream) {
    (void)in_sizes; (void)n_in; (void)d_ws; (void)ws_size; (void)out_size;
    const float* x = (const float*)d_in[0];
    float* out = (float*)d_out;
    // (16 batch * 3 ch) images of 512x512; tiles of 16 rows x 128 cols.
    dim3 grid(IMG_W / TILE_W, IMG_H / TILE_H, 16 * 3);
    dim3 block(256, 1, 1);
    local_variance_kernel<<<grid, block, 0, stream>>>(x, out);
}